// AttentionCTCLoss_23467701305800
// MI455X (gfx1250) — compile-verified
//
#include <hip/hip_runtime.h>

#define NB 64
#define NT 2000
#define NK 400
#define NS 801            // 2*K+1 extended CTC states
#define NEGF (-1e30f)
#define LOG2E_F 1.4426950408889634f
#define LN2_D   0.6931471805599453

// CDNA5 async global->LDS path (gfx1250): guard so host pass / older toolchains fall back.
#if defined(__AMDGCN__) && __has_builtin(__builtin_amdgcn_global_load_async_to_lds_b128) && __has_builtin(__builtin_amdgcn_s_wait_asynccnt)
#define USE_ASYNC_LDS 1
typedef int v4i_cdna5 __attribute__((ext_vector_type(4)));
typedef __attribute__((address_space(1))) v4i_cdna5 g_v4i;   // global (printed as __device__)
typedef __attribute__((address_space(3))) v4i_cdna5 l_v4i;   // LDS
#endif

__device__ __forceinline__ float fexp2(float x) {
#if __has_builtin(__builtin_amdgcn_exp2f)
  return __builtin_amdgcn_exp2f(x);
#else
  return exp2f(x);
#endif
}
__device__ __forceinline__ float flog2(float x) {
#if __has_builtin(__builtin_amdgcn_logf)
  return __builtin_amdgcn_logf(x);   // v_log_f32 is log2
#else
  return log2f(x);
#endif
}

// logaddexp in log2 domain, 2-way (uses 1 exp2 + 1 log2)
__device__ __forceinline__ float lse2_2(float a, float b) {
  float m = fmaxf(a, b);
  float d = -fabsf(a - b);
  return m + flog2(1.0f + fexp2(d));
}
// 3-way (3 parallel exp2 + 1 log2)
__device__ __forceinline__ float lse3_2(float a, float b, float c) {
  float m = fmaxf(fmaxf(a, b), c);
  float s = fexp2(a - m) + fexp2(b - m) + fexp2(c - m);
  return m + flog2(s);
}

// ---------------------------------------------------------------------------
// Kernel 1: lse[b,t] = log(sum over allowed classes of exp(lp))  (natural log)
// classes: blank (-1.0) + attn_logprob[b,0,t,0..in_len-1]. One wave per row.
// ---------------------------------------------------------------------------
__global__ __launch_bounds__(256) void lse_kernel(
    const float* __restrict__ attn_lp, const int* __restrict__ in_lens,
    float* __restrict__ lse) {
  const int row  = blockIdx.x * 8 + (threadIdx.x >> 5);
  const int lane = threadIdx.x & 31;
  if (row >= NB * NT) return;
  const int b = row / NT;
  const int in_len = in_lens[b];
  const float* r = attn_lp + (size_t)row * NK;

  // online logsumexp; blank seeded into lane 0
  float m = (lane == 0) ? -1.0f : NEGF;
  float s = (lane == 0) ?  1.0f : 0.0f;
  for (int j = lane; j < in_len; j += 32) {
    float x = r[j];
    if (x > m) { s = s * __expf(m - x) + 1.0f; m = x; }
    else       { s += __expf(x - m); }
  }
  for (int off = 16; off; off >>= 1) {
    float m2 = __shfl_xor(m, off);
    float s2 = __shfl_xor(s, off);
    float M  = fmaxf(m, m2);
    s = s * __expf(m - M) + s2 * __expf(m2 - M);
    m = M;
  }
  if (lane == 0) lse[row] = m + __logf(s);
}

// ---------------------------------------------------------------------------
// Kernel 2: per-sample alpha recursion on RAW logits in log2 domain.
// One block (832 thr = 26 waves) per sample; 1 state per thread; alpha is
// double-buffered in LDS (one barrier per step); attn row double-buffered via
// CDNA5 async global->LDS loads. True alpha = raw_alpha*ln2 - sum(lse[0..t]).
// ---------------------------------------------------------------------------
__global__ __launch_bounds__(832) void ctc_alpha_kernel(
    const float* __restrict__ attn_lp, const int* __restrict__ in_lens,
    const int* __restrict__ out_lens, const float* __restrict__ lse,
    float* __restrict__ loss_out) {
  __shared__ float aBuf[2][NS + 33];                       // [2][834]; state s at [2+s]
  __shared__ float rowBuf[2][NK] __attribute__((aligned(16)));
  __shared__ double wsum[26];

  const int b   = blockIdx.x;
  const int tid = threadIdx.x;
  const int s   = tid;
  const int in_len  = in_lens[b];
  const int out_len = out_lens[b];
  const int smax = 2 * in_len;                             // states above never matter
  const float* rows  = attn_lp + (size_t)b * NT * NK;
  const float* lse_b = lse + (size_t)b * NT;

  // t=0 init (raw lp, log2 domain): alpha[0]=blank(-1), alpha[1]=label1
  float a0;
  if (s == 0)      a0 = -LOG2E_F;
  else if (s == 1) a0 = rows[0] * LOG2E_F;
  else             a0 = NEGF;
  aBuf[1][2 + tid] = a0;
  if (tid < 2) { aBuf[0][tid] = NEGF; aBuf[1][tid] = NEGF; }

  // prefetch row t=1 into rowBuf[1]
#ifdef USE_ASYNC_LDS
  if (tid < NK / 4)
    __builtin_amdgcn_global_load_async_to_lds_b128(
        (g_v4i*)(rows + NK + tid * 4),
        (l_v4i*)(&rowBuf[1][tid * 4]), 0, 0);
#else
  if (tid < NK / 4) {
    float4 v = *(const float4*)(rows + NK + tid * 4);
    *(float4*)(&rowBuf[1][tid * 4]) = v;
  }
#endif

  const bool active = (s <= smax);
  const bool odd    = (s & 1) != 0;
  const int  cls    = (s - 1) >> 1;

  for (int t = 1; t < out_len; ++t) {
    const int p = t & 1;
#ifdef USE_ASYNC_LDS
    __builtin_amdgcn_s_wait_asynccnt(0);   // row t fully landed (this wave's part)
#endif
    __syncthreads();                       // all parts visible; alpha(t-1) published

    // prefetch row t+1 into the other buffer (overlaps with compute below)
#ifdef USE_ASYNC_LDS
    if (t + 1 < out_len && tid < NK / 4)
      __builtin_amdgcn_global_load_async_to_lds_b128(
          (g_v4i*)(rows + (size_t)(t + 1) * NK + tid * 4),
          (l_v4i*)(&rowBuf[p ^ 1][tid * 4]), 0, 0);
#else
    float4 nxt;
    const bool pre = (t + 1 < out_len) && (tid < NK / 4);
    if (pre) nxt = *(const float4*)(rows + (size_t)(t + 1) * NK + tid * 4);
#endif

    if (active) {
      float lp2 = odd ? rowBuf[p][cls] * LOG2E_F : -LOG2E_F;
      float a1  = aBuf[p][2 + s - 1];                      // s==0 reads NEG pad
      float nv;
      if (odd && s >= 3) {                                 // skip transition allowed
        float a2 = aBuf[p][s];                             // == [2 + s - 2]
        nv = lp2 + lse3_2(a0, a1, a2);
      } else {
        nv = lp2 + lse2_2(a0, a1);
      }
      a0 = nv;
      aBuf[p ^ 1][2 + s] = nv;                             // publish for next step
    }
#ifndef USE_ASYNC_LDS
    if (pre) *(float4*)(&rowBuf[p ^ 1][tid * 4]) = nxt;
#endif
  }

  __syncthreads();

  // total normalizer shift = sum lse[b, 0..out_len-1] (fp64 for accuracy)
  double part = 0.0;
  for (int t = tid; t < out_len; t += 832) part += (double)lse_b[t];
  for (int off = 16; off; off >>= 1) part += __shfl_xor(part, off);
  if ((tid & 31) == 0) wsum[tid >> 5] = part;
  __syncthreads();

  if (tid == 0) {
    double shift = 0.0;
    for (int w = 0; w < 26; ++w) shift += wsum[w];
    const int pf = out_len & 1;                            // buffer holding final alpha
    float aL = aBuf[pf][2 + smax];
    float aP = aBuf[pf][2 + smax - 1];
    float ll2 = lse2_2(aL, aP);
    float ll  = (float)((double)ll2 * LN2_D - shift);      // back to natural log, normalized
    float loss = (ll > 0.5f * NEGF) ? (-ll / (float)in_len) : 0.0f;
    loss_out[b] = loss;
  }
}

// ---------------------------------------------------------------------------
// Kernel 3: mean over the 64 per-sample losses (deterministic single-wave sum)
// ---------------------------------------------------------------------------
__global__ __launch_bounds__(32) void finalize_kernel(
    const float* __restrict__ loss, float* __restrict__ out) {
  int l = threadIdx.x;
  float v = loss[l] + loss[l + 32];
  for (int off = 16; off; off >>= 1) v += __shfl_xor(v, off);
  if (l == 0) out[0] = v / (float)NB;
}

extern "C" void kernel_launch(void* const* d_in, const int* in_sizes, int n_in,
                              void* d_out, int out_size, void* d_ws, size_t ws_size,
                              hipStream_t stream) {
  (void)in_sizes; (void)n_in; (void)out_size; (void)ws_size;
  // setup_inputs order: attn (unused), in_lens, out_lens, attn_logprob
  const int*   in_lens  = (const int*)d_in[1];
  const int*   out_lens = (const int*)d_in[2];
  const float* attn_lp  = (const float*)d_in[3];

  float* ws_lse  = (float*)d_ws;             // NB*NT floats
  float* ws_loss = ws_lse + (size_t)NB * NT; // NB floats

  lse_kernel<<<(NB * NT) / 8, 256, 0, stream>>>(attn_lp, in_lens, ws_lse);
  ctc_alpha_kernel<<<NB, 832, 0, stream>>>(attn_lp, in_lens, out_lens, ws_lse, ws_loss);
  finalize_kernel<<<1, 32, 0, stream>>>(ws_loss, (float*)d_out);
}